// GCNNet_37778532336372
// MI455X (gfx1250) — compile-verified
//
#include <hip/hip_runtime.h>
#include <hip/hip_bf16.h>

typedef __attribute__((ext_vector_type(2))) float v2f;
typedef __attribute__((ext_vector_type(8))) float v8f;

__device__ __forceinline__ void atomic_fadd(float* p, float v) {
    __hip_atomic_fetch_add(p, v, __ATOMIC_RELAXED, __HIP_MEMORY_SCOPE_AGENT);
}

// ---------------- degree / normalization ----------------
__global__ void k_deg_init(float* __restrict__ d1, float* __restrict__ d2, int n) {
    int i = blockIdx.x * blockDim.x + threadIdx.x;
    if (i < n) { d1[i] = 1.0f; d2[i] = 1.0f; }   // +1 = self loop
}

__global__ void k_deg_edges(const int* __restrict__ ei,
                            const unsigned char* __restrict__ m1,
                            const unsigned char* __restrict__ m2,
                            float* __restrict__ deg1, float* __restrict__ deg2, int nE) {
    int e = blockIdx.x * blockDim.x + threadIdx.x;
    if (e >= nE) return;
    int d = ei[nE + e];                          // edge_index[1][e]
    if (m1[e]) atomic_fadd(&deg1[d], 1.0f);
    if (m2[e]) atomic_fadd(&deg2[d], 1.0f);
}

__global__ void k_rsqrt(float* __restrict__ d1, float* __restrict__ d2, int n) {
    int i = blockIdx.x * blockDim.x + threadIdx.x;
    if (i < n) { d1[i] = rsqrtf(d1[i]); d2[i] = rsqrtf(d2[i]); }
}

__global__ void k_zero(float* __restrict__ p, size_t n) {
    size_t i = (size_t)blockIdx.x * blockDim.x + threadIdx.x;
    if (i < n) p[i] = 0.0f;
}

// ---------------- WMMA fp32 GEMM: C[n x NF] = A[n x K] @ B[K x NF] (+bias) ----------------
// One wave computes one 16x16 output tile. Wave w of a block handles column tile w.
// A layout (16x4 f32): lanes 0-15 -> M=lane, K={k,k+1}; lanes 16-31 -> M=lane-16, K={k+2,k+3}
// B layout (4x16 f32): vgpr0 lanes 0-15 -> K=k,N=lane | lanes 16-31 -> K=k+2; vgpr1 -> K=k+1 / k+3
// D layout (16x16 f32): vgpr i lanes 0-15 -> M=i,N=lane | lanes 16-31 -> M=i+8,N=lane-16
template<int K, int NF, bool BIAS>
__global__ void k_gemm_wmma(const float* __restrict__ A, const float* __restrict__ B,
                            const float* __restrict__ bias, float* __restrict__ C, int nRows) {
    const int lane = threadIdx.x & 31;
    const int wv   = threadIdx.x >> 5;
    const int row0 = blockIdx.x * 16;
    const int col0 = wv * 16;
    const int half = lane >> 4;          // 0 or 1
    const int l15  = lane & 15;
    const int kh   = half * 2;           // K sub-offset within a k-step of 4

    int arow = row0 + l15; if (arow > nRows - 1) arow = nRows - 1;  // clamp tail reads
    const float* __restrict__ Arow = A + (size_t)arow * K;
    const float* __restrict__ Bp   = B + col0 + l15;

    v8f acc0 = {}; v8f acc1 = {};
#pragma unroll
    for (int k = 0; k < K; k += 8) {
        v2f a0 = *(const v2f*)(Arow + k + kh);                       // 8B-aligned
        v2f b0; b0.x = Bp[(size_t)(k + kh) * NF];
                b0.y = Bp[(size_t)(k + kh + 1) * NF];
        acc0 = __builtin_amdgcn_wmma_f32_16x16x4_f32(false, a0, false, b0,
                                                     (short)0, acc0, false, false);
        v2f a1 = *(const v2f*)(Arow + k + 4 + kh);
        v2f b1; b1.x = Bp[(size_t)(k + 4 + kh) * NF];
                b1.y = Bp[(size_t)(k + 5 + kh) * NF];
        acc1 = __builtin_amdgcn_wmma_f32_16x16x4_f32(false, a1, false, b1,
                                                     (short)0, acc1, false, false);
    }

    float bcol = BIAS ? bias[col0 + l15] : 0.0f;
    const int mbase = row0 + half * 8;
#pragma unroll
    for (int i = 0; i < 8; ++i) {
        int r = mbase + i;
        if (r < nRows)
            C[(size_t)r * NF + col0 + l15] = acc0[i] + acc1[i] + bcol;
    }
}

// ---------------- edge scatter: AGG[dst] += H[src] * dis[src]*dis[dst] ----------------
template<int F>   // F == 128: one wave per edge, one float4 per lane
__global__ void k_scatter(const int* __restrict__ ei, const unsigned char* __restrict__ mask,
                          const float* __restrict__ dis, const float* __restrict__ H,
                          float* __restrict__ AGG, int nE) {
    int e = blockIdx.x * (blockDim.x >> 5) + (threadIdx.x >> 5);
    if (e >= nE) return;
    if (!mask[e]) return;                        // uniform per wave
    int s = ei[e];
    int d = ei[nE + e];
    float c = dis[s] * dis[d];
    int lane = threadIdx.x & 31;
    float4 v = ((const float4*)(H + (size_t)s * F))[lane];   // coalesced b128 gather
    float* ap = AGG + (size_t)d * F + lane * 4;
    atomic_fadd(ap + 0, v.x * c);
    atomic_fadd(ap + 1, v.y * c);
    atomic_fadd(ap + 2, v.z * c);
    atomic_fadd(ap + 3, v.w * c);
}

// ---------------- fused self-loop + bias + ReLU (in place on AGG) ----------------
template<int F>
__global__ void k_post(float* __restrict__ AGG, const float* __restrict__ H,
                       const float* __restrict__ dis, const float* __restrict__ b, int n) {
    size_t i = (size_t)blockIdx.x * blockDim.x + threadIdx.x;
    if (i >= (size_t)n * F) return;
    int node = (int)(i >> 7);                    // F == 128
    int f    = (int)(i & (F - 1));
    float s = dis[node];
    float v = AGG[i] + H[i] * (s * s) + b[f];
    AGG[i] = fmaxf(v, 0.0f);
}

extern "C" void kernel_launch(void* const* d_in, const int* in_sizes, int n_in,
                              void* d_out, int out_size, void* d_ws, size_t ws_size,
                              hipStream_t stream) {
    constexpr int IN = 128, HID = 128, OF = 128, OUTF = 64;

    const float*         x  = (const float*)d_in[0];
    const int*           ei = (const int*)d_in[1];
    const unsigned char* m1 = (const unsigned char*)d_in[2];
    const unsigned char* m2 = (const unsigned char*)d_in[3];
    const float*         W1 = (const float*)d_in[4];
    const float*         b1 = (const float*)d_in[5];
    const float*         W2 = (const float*)d_in[6];
    const float*         b2 = (const float*)d_in[7];
    const float*         Wl = (const float*)d_in[8];
    const float*         bl = (const float*)d_in[9];
    float* out = (float*)d_out;

    const int n  = in_sizes[0] / IN;     // 50000
    const int nE = in_sizes[1] / 2;      // 800000

    // workspace: dis1 | dis2 | bufH (n*128) | bufAgg (n*128)  ~= 51.6 MB
    float* dis1   = (float*)d_ws;
    float* dis2   = dis1 + n;
    float* bufH   = dis2 + n;
    float* bufAgg = bufH + (size_t)n * HID;

    const int T = 256;
    const int rowTiles = (n + 15) / 16;
    const size_t nf = (size_t)n * HID;
    const int nfBlocks = (int)((nf + T - 1) / T);
    const int edgeBlocks = (nE + 7) / 8;          // 8 waves (edges) per 256-thread block

    // degrees -> dis = rsqrt(deg)
    k_deg_init <<<(n + T - 1) / T, T, 0, stream>>>(dis1, dis2, n);
    k_deg_edges<<<(nE + T - 1) / T, T, 0, stream>>>(ei, m1, m2, dis1, dis2, nE);
    k_rsqrt    <<<(n + T - 1) / T, T, 0, stream>>>(dis1, dis2, n);

    // ---- layer 1: H = x @ W1 ; AGG = scatter ; AGG = relu(AGG + H*dis^2 + b1)
    k_gemm_wmma<IN, HID, false><<<rowTiles, 32 * (HID / 16), 0, stream>>>(x, W1, nullptr, bufH, n);
    k_zero     <<<nfBlocks, T, 0, stream>>>(bufAgg, nf);
    k_scatter<HID><<<edgeBlocks, 256, 0, stream>>>(ei, m1, dis1, bufH, bufAgg, nE);
    k_post<HID>   <<<nfBlocks, T, 0, stream>>>(bufAgg, bufH, dis1, b1, n);

    // ---- layer 2: H = out1 @ W2 ; scatter ; relu(...)
    k_gemm_wmma<HID, OF, false><<<rowTiles, 32 * (OF / 16), 0, stream>>>(bufAgg, W2, nullptr, bufH, n);
    k_zero     <<<nfBlocks, T, 0, stream>>>(bufAgg, nf);
    k_scatter<OF><<<edgeBlocks, 256, 0, stream>>>(ei, m2, dis2, bufH, bufAgg, nE);
    k_post<OF>   <<<nfBlocks, T, 0, stream>>>(bufAgg, bufH, dis2, b2, n);

    // ---- final linear: out = out2 @ Wl + bl
    k_gemm_wmma<OF, OUTF, true><<<rowTiles, 32 * (OUTF / 16), 0, stream>>>(bufAgg, Wl, bl, out, n);
}